// GATRegressor_38508676776274
// MI455X (gfx1250) — compile-verified
//
#include <hip/hip_runtime.h>
#include <math.h>

#define N_NODES   50000
#define N_EDGES   800000
#define E_TOT     (N_EDGES + N_NODES)   // reference appends self-loops
#define NUM_GRAPHS 64
#define NEG_SLOPE 0.2f
#define BLK 256

typedef __attribute__((ext_vector_type(2))) float v2f;
typedef __attribute__((ext_vector_type(8))) float v8f;

// ---------------- utility ----------------

__global__ void fill_kernel(float* __restrict__ p, float v, int n) {
  int i = blockIdx.x * blockDim.x + threadIdx.x;
  if (i < n) p[i] = v;
}

__device__ __forceinline__ float leaky(float v) { return v > 0.f ? v : NEG_SLOPE * v; }

__device__ __forceinline__ void edge_nodes(const int* __restrict__ ei, int e, int& s, int& d) {
  if (e < N_EDGES) { s = ei[e]; d = ei[N_EDGES + e]; }
  else             { s = e - N_EDGES; d = s; }        // self-loop
}

__device__ __forceinline__ void atomic_max_f32(float* addr, float v) {
  // sign-aware int punning: maps to global_atomic_max_i32 / global_atomic_min_u32
  if (v >= 0.f) atomicMax((int*)addr, __float_as_int(v));
  else          atomicMin((unsigned int*)addr, __float_as_uint(v));
}

__device__ __forceinline__ void atomic_add_f32(float* addr, float v) {
  __hip_atomic_fetch_add(addr, v, __ATOMIC_RELAXED, __HIP_MEMORY_SCOPE_AGENT);
}

// ---------------- dense projection: C[M,N] = A[M,K] @ B[K,N] via WMMA f32 16x16x4 ----------------
// One wave32 computes one 16x16 tile. M,N multiples of 16; K multiple of 4.
// A-layout (ISA 7.12.2, 32-bit A 16x4): lane holds row (lane&15), k-pair (lane>>4)*2.
// B-layout (4x16, K-striped like A's K): lane holds col (lane&15), same k-pair.
// C/D: VGPR v -> row v + 8*(lane>>4), col (lane&15).
__global__ void gemm_wmma_kernel(const float* __restrict__ A, const float* __restrict__ B,
                                 float* __restrict__ C, int M, int N, int K) {
  int tid  = blockIdx.x * blockDim.x + threadIdx.x;
  int wid  = tid >> 5;
  int lane = threadIdx.x & 31;
  int tiles_n = N >> 4;
  int tiles   = (M >> 4) * tiles_n;
  if (wid >= tiles) return;                    // whole-wave uniform exit (EXEC stays all-1s)
  int tm = wid / tiles_n, tn = wid % tiles_n;

  int row_a = tm * 16 + (lane & 15);
  int kgrp  = (lane >> 4) * 2;
  int col_b = tn * 16 + (lane & 15);

  const float* arow = A + (size_t)row_a * K + kgrp;
  v8f acc = {};
  for (int k = 0; k < K; k += 4) {
    v2f a = *(const v2f*)(arow + k);           // 8B-aligned contiguous pair
    v2f b;
    b.x = B[(size_t)(k + kgrp)     * N + col_b];
    b.y = B[(size_t)(k + kgrp + 1) * N + col_b];
    acc = __builtin_amdgcn_wmma_f32_16x16x4_f32(false, a, false, b, (short)0, acc, false, false);
  }

  int row_c = tm * 16 + (lane >> 4) * 8;
  int col_c = tn * 16 + (lane & 15);
#pragma unroll
  for (int v = 0; v < 8; ++v)
    C[(size_t)(row_c + v) * N + col_c] = acc[v];
}

// ---------------- per-(node,head) attention coefficients: dot(h, a_src/a_dst) over C=64 ----------------
__global__ void attn_kernel(const float* __restrict__ h, const float* __restrict__ a_src,
                            const float* __restrict__ a_dst, float* __restrict__ out_s,
                            float* __restrict__ out_d, int H) {
  int tid = blockIdx.x * blockDim.x + threadIdx.x;
  int wid = tid >> 5, lane = threadIdx.x & 31;
  if (wid >= N_NODES * H) return;
  int n = wid / H, hh = wid % H;
  const float* hp = h + (size_t)n * H * 64 + hh * 64;
  float h0 = hp[lane], h1 = hp[lane + 32];
  float s = h0 * a_src[hh * 64 + lane] + h1 * a_src[hh * 64 + lane + 32];
  float d = h0 * a_dst[hh * 64 + lane] + h1 * a_dst[hh * 64 + lane + 32];
#pragma unroll
  for (int m = 16; m >= 1; m >>= 1) {
    s += __shfl_xor(s, m, 32);
    d += __shfl_xor(d, m, 32);
  }
  if (lane == 0) { out_s[wid] = s; out_d[wid] = d; }
}

// ---------------- segment softmax pass 1: per-dst max ----------------
__global__ void edge_max_kernel(const int* __restrict__ ei, const float* __restrict__ asrc,
                                const float* __restrict__ adst, float* __restrict__ emax, int H) {
  int t = blockIdx.x * blockDim.x + threadIdx.x;
  if (t >= E_TOT * H) return;
  int e = t / H, hh = t % H;
  int s, d; edge_nodes(ei, e, s, d);
  float v = leaky(asrc[s * H + hh] + adst[d * H + hh]);
  atomic_max_f32(&emax[d * H + hh], v);
}

// ---------------- segment softmax pass 2: per-dst sum of exp ----------------
__global__ void edge_sum_kernel(const int* __restrict__ ei, const float* __restrict__ asrc,
                                const float* __restrict__ adst, const float* __restrict__ emax,
                                float* __restrict__ den, int H) {
  int t = blockIdx.x * blockDim.x + threadIdx.x;
  if (t >= E_TOT * H) return;
  int e = t / H, hh = t % H;
  int s, d; edge_nodes(ei, e, s, d);
  float v = leaky(asrc[s * H + hh] + adst[d * H + hh]);
  atomic_add_f32(&den[d * H + hh], __expf(v - emax[d * H + hh]));
}

// ---------------- pass 3: weighted aggregation, one wave per (edge, head) ----------------
__global__ void edge_agg_kernel(const int* __restrict__ ei, const float* __restrict__ asrc,
                                const float* __restrict__ adst, const float* __restrict__ emax,
                                const float* __restrict__ den, const float* __restrict__ hin,
                                float* __restrict__ agg, int H) {
  int tid = blockIdx.x * blockDim.x + threadIdx.x;
  int wid = tid >> 5, lane = threadIdx.x & 31;
  if (wid >= E_TOT * H) return;
  int e = wid / H, hh = wid % H;
  int s, d; edge_nodes(ei, e, s, d);
  float v = leaky(asrc[s * H + hh] + adst[d * H + hh]);
  float coeff = __expf(v - emax[d * H + hh]) / (den[d * H + hh] + 1e-16f);
  const float* hs = hin + (size_t)s * H * 64 + hh * 64;
  float*       ad = agg + (size_t)d * H * 64 + hh * 64;
  atomic_add_f32(&ad[lane],      coeff * hs[lane]);
  atomic_add_f32(&ad[lane + 32], coeff * hs[lane + 32]);
}

// ---------------- bias + ELU (F is a power of two) ----------------
__global__ void bias_elu_kernel(const float* __restrict__ in, const float* __restrict__ bias,
                                float* __restrict__ out, int n, int F) {
  int i = blockIdx.x * blockDim.x + threadIdx.x;
  if (i >= n) return;
  float v = in[i] + bias[i & (F - 1)];
  out[i] = v > 0.f ? v : (__expf(v) - 1.f);
}

// ---------------- global mean pool (sums + counts), one wave per node ----------------
__global__ void pool_kernel(const float* __restrict__ h, const int* __restrict__ batch,
                            float* __restrict__ sums, float* __restrict__ cnts) {
  int tid = blockIdx.x * blockDim.x + threadIdx.x;
  int wid = tid >> 5, lane = threadIdx.x & 31;
  if (wid >= N_NODES) return;
  int g = batch[wid];
  const float* hp = h + (size_t)wid * 64;
  atomic_add_f32(&sums[g * 64 + lane],      hp[lane]);
  atomic_add_f32(&sums[g * 64 + lane + 32], hp[lane + 32]);
  if (lane == 0) atomic_add_f32(&cnts[g], 1.0f);
}

// ---------------- tiny head MLP: [G,64] -> relu@[64,32] -> @[32,1] ----------------
__global__ void mlp_kernel(const float* __restrict__ sums, const float* __restrict__ cnts,
                           const float* __restrict__ lw1, const float* __restrict__ lb1,
                           const float* __restrict__ lw2, const float* __restrict__ lb2,
                           float* __restrict__ out) {
  int g = threadIdx.x;
  if (g >= NUM_GRAPHS) return;
  float inv = 1.f / fmaxf(cnts[g], 1.f);
  float acc = 0.f;
  for (int j = 0; j < 32; ++j) {
    float hv = lb1[j];
    for (int c = 0; c < 64; ++c)
      hv += sums[g * 64 + c] * inv * lw1[c * 32 + j];
    acc += fmaxf(hv, 0.f) * lw2[j];
  }
  out[g] = acc + lb2[0];
}

// ---------------- host launcher ----------------
extern "C" void kernel_launch(void* const* d_in, const int* in_sizes, int n_in,
                              void* d_out, int out_size, void* d_ws, size_t ws_size,
                              hipStream_t stream) {
  const float* x    = (const float*)d_in[0];
  const int*   ei   = (const int*)d_in[1];
  const int*   batch= (const int*)d_in[2];
  const float* W1   = (const float*)d_in[3];
  const float* as1  = (const float*)d_in[4];
  const float* ad1  = (const float*)d_in[5];
  const float* b1   = (const float*)d_in[6];
  const float* W2   = (const float*)d_in[7];
  const float* as2  = (const float*)d_in[8];
  const float* ad2  = (const float*)d_in[9];
  const float* b2   = (const float*)d_in[10];
  const float* lw1  = (const float*)d_in[11];
  const float* lb1  = (const float*)d_in[12];
  const float* lw2  = (const float*)d_in[13];
  const float* lb2  = (const float*)d_in[14];
  float* out = (float*)d_out;

  char* ws = (char*)d_ws;
  size_t off = 0;
  auto alloc = [&](size_t bytes) {
    char* p = ws + off;
    off += (bytes + 255) & ~(size_t)255;
    return (float*)p;
  };
  float* h1    = alloc((size_t)N_NODES * 256 * 4); // layer1 projection; reused for ELU output
  float* agg1  = alloc((size_t)N_NODES * 256 * 4);
  float* asr1  = alloc((size_t)N_NODES * 4 * 4);
  float* adr1  = alloc((size_t)N_NODES * 4 * 4);
  float* emax1 = alloc((size_t)N_NODES * 4 * 4);
  float* den1  = alloc((size_t)N_NODES * 4 * 4);
  float* h2    = alloc((size_t)N_NODES * 64 * 4);  // layer2 projection; reused for ELU output
  float* agg2  = alloc((size_t)N_NODES * 64 * 4);
  float* asr2  = alloc((size_t)N_NODES * 4);
  float* adr2  = alloc((size_t)N_NODES * 4);
  float* emax2 = alloc((size_t)N_NODES * 4);
  float* den2  = alloc((size_t)N_NODES * 4);
  float* sums  = alloc((size_t)NUM_GRAPHS * 64 * 4);
  float* cnts  = alloc((size_t)NUM_GRAPHS * 4);
  (void)ws_size; (void)in_sizes; (void)n_in; (void)out_size;

  auto cdiv = [](long long a, long long b) { return (int)((a + b - 1) / b); };

  // ---------- Layer 1: GATConv(128 -> 4 heads x 64) ----------
  {
    const int H = 4;
    int tiles = (N_NODES / 16) * (256 / 16);
    gemm_wmma_kernel<<<cdiv((long long)tiles * 32, BLK), BLK, 0, stream>>>(x, W1, h1, N_NODES, 256, 128);
    attn_kernel<<<cdiv((long long)N_NODES * H * 32, BLK), BLK, 0, stream>>>(h1, as1, ad1, asr1, adr1, H);
    fill_kernel<<<cdiv(N_NODES * H, BLK), BLK, 0, stream>>>(emax1, -INFINITY, N_NODES * H);
    fill_kernel<<<cdiv(N_NODES * H, BLK), BLK, 0, stream>>>(den1, 0.f, N_NODES * H);
    fill_kernel<<<cdiv((long long)N_NODES * 256, BLK), BLK, 0, stream>>>(agg1, 0.f, N_NODES * 256);
    edge_max_kernel<<<cdiv((long long)E_TOT * H, BLK), BLK, 0, stream>>>(ei, asr1, adr1, emax1, H);
    edge_sum_kernel<<<cdiv((long long)E_TOT * H, BLK), BLK, 0, stream>>>(ei, asr1, adr1, emax1, den1, H);
    edge_agg_kernel<<<cdiv((long long)E_TOT * H * 32, BLK), BLK, 0, stream>>>(ei, asr1, adr1, emax1, den1, h1, agg1, H);
    bias_elu_kernel<<<cdiv((long long)N_NODES * 256, BLK), BLK, 0, stream>>>(agg1, b1, h1, N_NODES * 256, 256);
  }

  // ---------- Layer 2: GATConv(256 -> 1 head x 64) ----------
  {
    const int H = 1;
    int tiles = (N_NODES / 16) * (64 / 16);
    gemm_wmma_kernel<<<cdiv((long long)tiles * 32, BLK), BLK, 0, stream>>>(h1, W2, h2, N_NODES, 64, 256);
    attn_kernel<<<cdiv((long long)N_NODES * H * 32, BLK), BLK, 0, stream>>>(h2, as2, ad2, asr2, adr2, H);
    fill_kernel<<<cdiv(N_NODES * H, BLK), BLK, 0, stream>>>(emax2, -INFINITY, N_NODES * H);
    fill_kernel<<<cdiv(N_NODES * H, BLK), BLK, 0, stream>>>(den2, 0.f, N_NODES * H);
    fill_kernel<<<cdiv((long long)N_NODES * 64, BLK), BLK, 0, stream>>>(agg2, 0.f, N_NODES * 64);
    edge_max_kernel<<<cdiv((long long)E_TOT * H, BLK), BLK, 0, stream>>>(ei, asr2, adr2, emax2, H);
    edge_sum_kernel<<<cdiv((long long)E_TOT * H, BLK), BLK, 0, stream>>>(ei, asr2, adr2, emax2, den2, H);
    edge_agg_kernel<<<cdiv((long long)E_TOT * H * 32, BLK), BLK, 0, stream>>>(ei, asr2, adr2, emax2, den2, h2, agg2, H);
    bias_elu_kernel<<<cdiv((long long)N_NODES * 64, BLK), BLK, 0, stream>>>(agg2, b2, h2, N_NODES * 64, 64);
  }

  // ---------- Pool + MLP head ----------
  fill_kernel<<<cdiv(NUM_GRAPHS * 64, BLK), BLK, 0, stream>>>(sums, 0.f, NUM_GRAPHS * 64);
  fill_kernel<<<1, BLK, 0, stream>>>(cnts, 0.f, NUM_GRAPHS);
  pool_kernel<<<cdiv((long long)N_NODES * 32, BLK), BLK, 0, stream>>>(h2, batch, sums, cnts);
  mlp_kernel<<<1, 64, 0, stream>>>(sums, cnts, lw1, lb1, lw2, lb2, out);
}